// RelationFeatureExtractor_9259949490893
// MI455X (gfx1250) — compile-verified
//
#include <hip/hip_runtime.h>
#include <hip/hip_bf16.h>

// ---------------------------------------------------------------------------
// RelationFeatureExtractor for MI455X (gfx1250)
//   masks:        (300, 150, 28, 28) f32
//   scores:       (300, 151)         f32
//   W_var:        (302, 6)           f32
//   b_var:        (6,)               f32
//   rel_pair_idx: (512, 2)           i32
//   out:          (512, 150, 28, 28) f32
//
// K1: per-pair variance (dot-302 x6) + 2x 7x7 normalized Gaussian kernels.
// K2: one workgroup per pair. Depthwise 7x7 conv as 7 banded Toeplitz GEMMs
//     (O = sum_dy I_shift(dy) * T_dy) on V_WMMA_F32_16X16X4_F32 (exact fp32),
//     band-aware K-chunk skipping, software-pipelined A-fragment DS loads,
//     GLOBAL_LOAD_ASYNC_TO_LDS_B128 for the mask tiles (ASYNCcnt),
//     head*tail product fused at the store.
// ---------------------------------------------------------------------------

typedef __attribute__((ext_vector_type(2))) float v2f;
typedef __attribute__((ext_vector_type(8))) float v8f;

#define KSIZE   7
#define HH      28        // image H == W
#define NCH     150       // channels
#define SROW    151       // scores row length (NUM_CLASSES+1)
#define PROW    38        // padded image rows: 3 + 28 + 7
#define IPF     (PROW*HH) // floats per padded image (1064)
#define TCOLS   32        // Toeplitz cols padded 28 -> 32
#define TFLOATS (2*KSIZE*HH*TCOLS)          // 12544 floats
#define NWAVES  8
#define LDS_FLOATS (TFLOATS + NWAVES*2*IPF) // 29568 floats
#define LDS_BYTES  (LDS_FLOATS*4)           // 118272 B (< 320KB WGP LDS)
#define CH16    (HH*HH/4)                   // 196 float4 chunks per channel

// ---------------------------------------------------------------------------
// Kernel 1: variance GEMM + gaussian kernel construction (one thread per pair)
// ---------------------------------------------------------------------------
__global__ void gauss_kernels_k(const float* __restrict__ scores,
                                const float* __restrict__ W_var,
                                const float* __restrict__ b_var,
                                const int*   __restrict__ rel,
                                float*       __restrict__ kout,
                                int E) {
  int e = blockIdx.x * blockDim.x + threadIdx.x;
  if (e >= E) return;
  int hi = rel[2 * e + 0];
  int ti = rel[2 * e + 1];

  float acc[6];
#pragma unroll
  for (int j = 0; j < 6; ++j) acc[j] = b_var[j];

  const float* sh = scores + (size_t)hi * SROW;
  const float* st = scores + (size_t)ti * SROW;
  for (int i = 0; i < SROW; ++i) {
    float s = sh[i];
#pragma unroll
    for (int j = 0; j < 6; ++j) acc[j] = fmaf(s, W_var[i * 6 + j], acc[j]);
  }
  for (int i = 0; i < SROW; ++i) {
    float s = st[i];
#pragma unroll
    for (int j = 0; j < 6; ++j) acc[j] = fmaf(s, W_var[(SROW + i) * 6 + j], acc[j]);
  }

  for (int sd = 0; sd < 2; ++sd) {
    float s0  = fmaxf(expf(0.5f * acc[sd * 3 + 0]), 1e-7f);
    float s1  = fmaxf(expf(0.5f * acc[sd * 3 + 1]), 1e-7f);
    float rho = tanhf(acc[sd * 3 + 2]);
    float omr = fminf(1.0f - rho * rho, 1.0f);
    float inv2 = 1.0f / (2.0f * omr);
    float inv_s0 = 1.0f / s0, inv_s1 = 1.0f / s1;

    float kv[KSIZE * KSIZE];
    float sum = 0.0f;
    for (int r = 0; r < KSIZE; ++r) {
      float yb = (float)(r - 3) * inv_s1;
      for (int c = 0; c < KSIZE; ++c) {
        float xb = (float)(c - 3) * inv_s0;
        float z  = (xb + yb) * (xb + yb) - 2.0f * xb * yb * (1.0f + rho);
        float v  = expf(-z * inv2);
        kv[r * KSIZE + c] = v;
        sum += v;
      }
    }
    float inv = 1.0f / sum;
    float* kd = kout + ((size_t)e * 2 + sd) * (KSIZE * KSIZE);
    for (int i = 0; i < KSIZE * KSIZE; ++i) kd[i] = kv[i] * inv;
  }
}

// ---------------------------------------------------------------------------
// Kernel 2 helpers
// ---------------------------------------------------------------------------

// Async copy one 28x28 channel image (196 aligned 16B chunks) from global
// memory directly into the padded LDS slot (rows 3..30). ASYNCcnt-tracked.
__device__ __forceinline__ void async_load_image(unsigned lds_base,
                                                 const float* __restrict__ src,
                                                 int lane) {
  for (int i = lane; i < CH16; i += 32) {
    unsigned loff = lds_base + 3u * HH * 4u + (unsigned)i * 16u;
    unsigned goff = (unsigned)i * 16u;
    asm volatile("global_load_async_to_lds_b128 %0, %1, %2"
                 :
                 : "v"(loff), "v"(goff), "s"(src)
                 : "memory");
  }
}

// Load the 7 A-fragments (f32 16x4 layout) of one dy row batch.
// A layout: lane l: M=l&15, VGPR0 holds K = k0+(l<16?0:2), VGPR1 = +1
__device__ __forceinline__ void load_afrags(v2f* __restrict__ afr,
                                            const float* __restrict__ iprow,
                                            int khalf) {
#pragma unroll
  for (int kc = 0; kc < 7; ++kc) {
    const int kk = kc * 4 + khalf;
    afr[kc].x = iprow[kk];
    afr[kc].y = iprow[kk + 1];
  }
}

// One side's 16-row x 28-col conv tile via banded-Toeplitz WMMA GEMMs,
// software-pipelined: dy+1 A-fragments load behind dy's WMMA chain.
// B layout (f32 4x16): VGPR0 holds rows K0|K2 (lanes 0-15|16-31), VGPR1 K1|K3
// C/D layout: VGPR r: M = r + (lane<16?0:8), N = lane&15
__device__ __forceinline__ void conv_side(const float* __restrict__ ip,
                                          const float* __restrict__ Ts,
                                          int lane, int y0,
                                          v8f& acc0, v8f& acc1) {
  const int m     = lane & 15;
  const int khalf = (lane < 16) ? 0 : 2;

  v2f cur[7], nxt[7];
  load_afrags(cur, ip + (y0 + 0 + m) * HH, khalf);

  for (int dy = 0; dy < KSIZE; ++dy) {
    if (dy < KSIZE - 1)
      load_afrags(nxt, ip + (y0 + dy + 1 + m) * HH, khalf);

    const float* Td = Ts + dy * HH * TCOLS;
#pragma unroll
    for (int kc = 0; kc < 7; ++kc) {
      const int k0 = kc * 4;
      const int kk = k0 + khalf;
      if (k0 <= 16) {  // band for output cols 0..15 touches t in [0,18]
        v2f b;
        b.x = Td[kk * TCOLS + m];
        b.y = Td[(kk + 1) * TCOLS + m];
        acc0 = __builtin_amdgcn_wmma_f32_16x16x4_f32(
            false, cur[kc], false, b, (short)0, acc0, false, false);
      }
      if (k0 >= 12) {  // band for output cols 16..27 touches t in [13,27]
        v2f b;
        b.x = Td[kk * TCOLS + 16 + m];
        b.y = Td[(kk + 1) * TCOLS + 16 + m];
        acc1 = __builtin_amdgcn_wmma_f32_16x16x4_f32(
            false, cur[kc], false, b, (short)0, acc1, false, false);
      }
    }
#pragma unroll
    for (int kc = 0; kc < 7; ++kc) cur[kc] = nxt[kc];
  }
}

// ---------------------------------------------------------------------------
// Kernel 2: one workgroup per relation pair; wave-per-channel depthwise conv
// (head & tail) fused with the elementwise product.
// ---------------------------------------------------------------------------
__global__ void __launch_bounds__(256)
union_conv_k(const float* __restrict__ masks,
             const int*   __restrict__ rel,
             const float* __restrict__ kern,
             float*       __restrict__ out) {
  extern __shared__ float smem[];
  float* T  = smem;            // [2][7][28][32] banded Toeplitz, zero padded
  float* IP = smem + TFLOATS;  // [NWAVES][2][PROW*28] per-wave image slots

  const int e    = blockIdx.x;
  const int tid  = threadIdx.x;
  const int lane = tid & 31;
  const int wave = tid >> 5;

  const int hi = rel[2 * e + 0];
  const int ti = rel[2 * e + 1];

  // Build Toeplitz matrices T_dy[t][x] = w[side][dy][t-x+3] (banded, 7 wide)
  const float* ke = kern + (size_t)e * 2 * (KSIZE * KSIZE);
  for (int i = tid; i < TFLOATS; i += 256) {
    int x  = i & (TCOLS - 1);
    int t  = (i >> 5) % HH;
    int dy = ((i >> 5) / HH) % KSIZE;
    int sd = i / (KSIZE * HH * TCOLS);
    int dx = t - x + 3;
    float wv = 0.0f;
    if (x < HH && dx >= 0 && dx < KSIZE) wv = ke[sd * 49 + dy * KSIZE + dx];
    T[i] = wv;
  }

  float* ipH = IP + (size_t)(wave * 2 + 0) * IPF;
  float* ipT = IP + (size_t)(wave * 2 + 1) * IPF;
  const unsigned ipH_lds = (unsigned)(uintptr_t)ipH;  // LDS byte offset
  const unsigned ipT_lds = (unsigned)(uintptr_t)ipT;

  // Zero the channel-invariant pad rows (0..2 and 31..37) once per slot.
  // 70 float4 per slot: 21 for top pad, 49 for bottom pad.
  const float4 z4 = make_float4(0.f, 0.f, 0.f, 0.f);
  for (int i = lane; i < 140; i += 32) {
    int slot = i / 70;
    int j    = i - slot * 70;
    int pr, c4;
    if (j < 21) { pr = j / 7;            c4 = (j % 7) * 4; }
    else        { int jj = j - 21; pr = 31 + jj / 7; c4 = (jj % 7) * 4; }
    float* dst = (slot == 0 ? ipH : ipT) + pr * HH + c4;
    *(float4*)dst = z4;
  }
  __syncthreads();

  const float* Th = T;
  const float* Tt = T + KSIZE * HH * TCOLS;
  const int m_add = (lane < 16) ? 0 : 8;
  const int n     = lane & 15;

  for (int c = wave; c < NCH; c += NWAVES) {
    const float* srcH = masks + ((size_t)hi * NCH + c) * (HH * HH);
    const float* srcT = masks + ((size_t)ti * NCH + c) * (HH * HH);

    // Direct global->LDS async DMA of both images, then one wait.
    async_load_image(ipH_lds, srcH, lane);
    async_load_image(ipT_lds, srcT, lane);
    if (c + NWAVES < NCH) {  // global_prefetch_b8 next channel into L2/L0
      __builtin_prefetch(srcH + NWAVES * HH * HH, 0, 1);
      __builtin_prefetch(srcT + NWAVES * HH * HH, 0, 1);
    }
    asm volatile("s_wait_asynccnt 0x0" ::: "memory");

    float* ob = out + ((size_t)e * NCH + c) * (HH * HH);
    for (int y0 = 0; y0 < HH; y0 += 16) {
      v8f aH0 = {}, aH1 = {}, aT0 = {}, aT1 = {};
      conv_side(ipH, Th, lane, y0, aH0, aH1);
      conv_side(ipT, Tt, lane, y0, aT0, aT1);
#pragma unroll
      for (int r = 0; r < 8; ++r) {
        int y = y0 + r + m_add;
        if (y < HH) {
          ob[y * HH + n] = aH0[r] * aT0[r];
          if (16 + n < HH) ob[y * HH + 16 + n] = aH1[r] * aT1[r];
        }
      }
    }
  }
}

// ---------------------------------------------------------------------------
// Launch
// ---------------------------------------------------------------------------
extern "C" void kernel_launch(void* const* d_in, const int* in_sizes, int n_in,
                              void* d_out, int out_size, void* d_ws, size_t ws_size,
                              hipStream_t stream) {
  const float* masks  = (const float*)d_in[0];
  const float* scores = (const float*)d_in[1];
  const float* W_var  = (const float*)d_in[2];
  const float* b_var  = (const float*)d_in[3];
  const int*   rel    = (const int*)d_in[4];
  float* out  = (float*)d_out;
  float* kbuf = (float*)d_ws;  // (E, 2, 49) gaussian kernels

  const int E = in_sizes[4] / 2;

  gauss_kernels_k<<<(E + 255) / 256, 256, 0, stream>>>(scores, W_var, b_var,
                                                       rel, kbuf, E);
  union_conv_k<<<E, 256, LDS_BYTES, stream>>>(masks, rel, kbuf, out);
}